// AdaptiveEncoderCls_91233695301835
// MI455X (gfx1250) — compile-verified
//
#include <hip/hip_runtime.h>
#include <math.h>

typedef float v8f __attribute__((ext_vector_type(8)));
typedef float v2f __attribute__((ext_vector_type(2)));

#define FINF __builtin_inff()

// ---------------------------------------------------------------- zero
__global__ void zero_kernel(float* __restrict__ p, int n) {
    int i = blockIdx.x * blockDim.x + threadIdx.x;
    if (i < n) p[i] = 0.f;
}

// ---------------------------------------------------- per-(b,c) stats over N
__global__ void colstats_kernel(const float* __restrict__ x, int N, float* __restrict__ gb) {
    int g = blockIdx.x; int b = g / 3, c = g % 3; int tid = threadIdx.x;
    __shared__ float rs[256], r2[256];
    float s = 0.f, s2 = 0.f;
    for (int n = tid; n < N; n += 256) {
        float v = x[((size_t)b * N + n) * 3 + c];
        s += v; s2 += v * v;
    }
    rs[tid] = s; r2[tid] = s2; __syncthreads();
    for (int st = 128; st; st >>= 1) {
        if (tid < st) { rs[tid] += rs[tid + st]; r2[tid] += r2[tid + st]; }
        __syncthreads();
    }
    if (tid == 0) { gb[g * 2 + 0] = rs[0]; gb[g * 2 + 1] = r2[0]; }
}

// gstd = mean over G groups of std(ddof=1, n samples); -> asig, blend
__global__ void params_kernel(const float* __restrict__ gb, int G, float n, float* __restrict__ prm) {
    if (threadIdx.x || blockIdx.x) return;
    float acc = 0.f;
    for (int g = 0; g < G; ++g) {
        float s = gb[g * 2], s2 = gb[g * 2 + 1];
        float var = (s2 - s * s / n) / (n - 1.f);
        acc += sqrtf(fmaxf(var, 0.f));
    }
    float gstd = acc / (float)G;
    prm[0] = 0.26f * (1.f + gstd);                       // asig
    prm[1] = 1.f / (1.f + expf(-(gstd - 0.1f) * 10.f));  // blend
}

// ------------------------------------------------- initial adaptive embed (dim 64)
__global__ void init_embed_kernel(const float* __restrict__ x, const float* __restrict__ prm,
                                  float* __restrict__ feat, size_t total /* B*N*64 */) {
    size_t i = (size_t)blockIdx.x * blockDim.x + threadIdx.x;
    if (i >= total) return;
    int j = (int)(i % 64); size_t bn = i / 64;
    int f = (j * 65) / 63;      // out_idx: linspace(0, 65, 64) truncated
    int c = f / 22, t = f % 22; // feat_dim = 22
    float fv = -1.f + 2.f * (float)(t + 1) / 23.f;
    float asig = prm[0], blend = prm[1];
    float z = (x[bn * 3 + c] - fv) / (asig + 1e-6f);
    feat[i] = blend * expf(-0.5f * z * z) + (1.f - blend) * cosf(z);
}

// -------------------------------------------------------------- FPS (sequential)
__global__ void __launch_bounds__(256) fps_kernel(const float* __restrict__ x,
                                                  int* __restrict__ fps_idx, int N, int S) {
    int b = blockIdx.x, tid = threadIdx.x;
    __shared__ float ldsDist[4096];
    __shared__ float redD[256]; __shared__ int redI[256];
    __shared__ float lastPt[3];
    for (int i = tid; i < N; i += 256) ldsDist[i] = 1e10f;
    if (tid == 0) {
        fps_idx[(size_t)b * S + 0] = 0;
        lastPt[0] = x[(size_t)b * N * 3 + 0];
        lastPt[1] = x[(size_t)b * N * 3 + 1];
        lastPt[2] = x[(size_t)b * N * 3 + 2];
    }
    __syncthreads();
    for (int it = 1; it < S; ++it) {
        float lx = lastPt[0], ly = lastPt[1], lz = lastPt[2];
        float bestD = -1.f; int bestI = 0;
        for (int i = tid; i < N; i += 256) {
            const float* p = &x[((size_t)b * N + i) * 3];
            float dx = p[0] - lx, dy = p[1] - ly, dz = p[2] - lz;
            float d = dx * dx + dy * dy + dz * dz;
            d = fminf(ldsDist[i], d);
            ldsDist[i] = d;
            if (d > bestD) { bestD = d; bestI = i; }  // ascending i => first max kept
        }
        redD[tid] = bestD; redI[tid] = bestI;
        __syncthreads();
        for (int s = 128; s; s >>= 1) {
            if (tid < s) {
                float od = redD[tid + s]; int oi = redI[tid + s];
                if (od > redD[tid] || (od == redD[tid] && oi < redI[tid])) { redD[tid] = od; redI[tid] = oi; }
            }
            __syncthreads();
        }
        if (tid == 0) {
            int nxt = redI[0];
            fps_idx[(size_t)b * S + it] = nxt;
            lastPt[0] = x[((size_t)b * N + nxt) * 3 + 0];
            lastPt[1] = x[((size_t)b * N + nxt) * 3 + 1];
            lastPt[2] = x[((size_t)b * N + nxt) * 3 + 2];
        }
        __syncthreads();
    }
}

// -------------------------------------------------------------- row gather
__global__ void gather_kernel(const float* __restrict__ src, const int* __restrict__ idx,
                              float* __restrict__ dst, int S, int C, int N, size_t total) {
    size_t i = (size_t)blockIdx.x * blockDim.x + threadIdx.x;
    if (i >= total) return;
    int c = (int)(i % C); size_t bs = i / C; int s = (int)(bs % S); int b = (int)(bs / S);
    dst[i] = src[((size_t)b * N + idx[(size_t)b * S + s]) * C + c];
}

// -------------------------------------------------------------- KNN with WMMA f32 16x16x4
// Block = 512 threads (16 waves); 16 queries per block. Distances via
// V_WMMA_F32_16X16X4_F32 (K=3 padded to 4), written to LDS; streaming top-32
// selection per query over 512-column chunks with a 32-entry sorted carry.
__global__ void __launch_bounds__(512) knn_kernel(const float* __restrict__ xq,
                                                  const float* __restrict__ xp,
                                                  int* __restrict__ knn_idx, int S, int N) {
    const int CH = 512, RL = 544;  // chunk columns, row length (CH + 32 carry)
    __shared__ float ldsD[16 * 544];   // 34.8 KB distance rows
    __shared__ float ldsCd[512];       // carry dists (16 x 32)
    __shared__ int   ldsCi[2][512];    // carry indices, double buffered
    __shared__ float ldsQ[64];         // 16 queries x 4 (z-padded)
    __shared__ float ldsQ2[16];        // |q|^2
    int b = blockIdx.y, qbase = blockIdx.x * 16;
    int tid = threadIdx.x, wave = tid >> 5, lane = tid & 31;
    int m = lane & 15, h = lane >> 4;

    if (tid < 16) {
        const float* q = &xq[((size_t)b * S + qbase + tid) * 3];
        float qx = q[0], qy = q[1], qz = q[2];
        ldsQ[tid * 4 + 0] = qx; ldsQ[tid * 4 + 1] = qy;
        ldsQ[tid * 4 + 2] = qz; ldsQ[tid * 4 + 3] = 0.f;
        ldsQ2[tid] = qx * qx + qy * qy + qz * qz;
    }
    ldsCd[tid] = FINF;
    ldsCi[1][tid] = 0x7fffffff;
    __syncthreads();

    // A-matrix (16x4 f32): lanes 0-15 hold K0,K1 of row m; lanes 16-31 hold K2,K3.
    v2f a; a.x = ldsQ[m * 4 + 2 * h + 0]; a.y = ldsQ[m * 4 + 2 * h + 1];
    float q2r[8];
#pragma unroll
    for (int v = 0; v < 8; v++) q2r[v] = ldsQ2[v + 8 * h];  // C/D rows: v + 8*half

    int nch = (N + CH - 1) / CH;
    for (int ci = 0; ci < nch; ++ci) {
        int base = ci * CH;
        int cols = N - base; if (cols > CH) cols = CH;
        int tiles = cols >> 4;
        // distance tiles: each wave handles tiles wave, wave+16, ...
        for (int t = wave; t < tiles; t += 16) {
            int col = base + t * 16 + m;
            const float* p = &xp[((size_t)b * N + col) * 3];
            float px = p[0], py = p[1], pz = p[2];
            v2f bb;  // B-matrix (4x16): lane col n=m; lanes 0-15 K0,K1; lanes 16-31 K2,K3
            if (h == 0) { bb.x = px; bb.y = py; } else { bb.x = pz; bb.y = 0.f; }
            float p2 = px * px + py * py + pz * pz;
            v8f acc = {0.f, 0.f, 0.f, 0.f, 0.f, 0.f, 0.f, 0.f};
            acc = __builtin_amdgcn_wmma_f32_16x16x4_f32(false, a, false, bb,
                                                        (short)0, acc, false, false);
#pragma unroll
            for (int v = 0; v < 8; v++)
                ldsD[(v + 8 * h) * RL + t * 16 + m] = q2r[v] + p2 - 2.f * acc[v];
        }
        if (cols < CH) {  // pad partial chunk with +inf
            int rem = CH - cols;
            for (int i = tid; i < 16 * rem; i += 512) {
                int row = i / rem, cc = cols + i % rem;
                ldsD[row * RL + cc] = FINF;
            }
        }
        { // append carry
            int row = tid >> 5, kk = tid & 31;
            ldsD[row * RL + CH + kk] = ldsCd[row * 32 + kk];
        }
        __syncthreads();

        int newSel = ci & 1, oldSel = newSel ^ 1;
        {   // extraction: wave `wave` owns query row `wave`; 32 ascending min-extracts
            int row = wave;
            volatile float* Dr = &ldsD[row * RL];
            const int* oldCi = &ldsCi[oldSel][row * 32];
            for (int k = 0; k < 32; ++k) {
                float bd = FINF; int bi = 0x7fffffff; int bp = 0;
                for (int pos = lane; pos < RL; pos += 32) {
                    float d = Dr[pos];
                    int gi = (pos < CH) ? (base + pos) : oldCi[pos - CH];
                    if (d < bd || (d == bd && gi < bi)) { bd = d; bi = gi; bp = pos; }
                }
#pragma unroll
                for (int off = 16; off; off >>= 1) {
                    float od = __shfl_xor(bd, off, 32);
                    int   oi = __shfl_xor(bi, off, 32);
                    int   op = __shfl_xor(bp, off, 32);
                    if (od < bd || (od == bd && oi < bi)) { bd = od; bi = oi; bp = op; }
                }
                if (lane == 0) {
                    ldsCd[row * 32 + k] = bd;
                    ldsCi[newSel][row * 32 + k] = bi;
                    Dr[bp] = FINF;
                }
            }
        }
        __syncthreads();
    }
    {   // write 32 ascending-distance neighbor indices per query
        int row = tid >> 5, kk = tid & 31;
        knn_idx[((size_t)b * S + qbase + row) * 32 + kk] = ldsCi[(nch - 1) & 1][row * 32 + kk];
    }
}

// --------------------------------------------- per-k diff stats (x and feat diffs)
__global__ void __launch_bounds__(128) diffstats_kernel(
        const float* __restrict__ x, const float* __restrict__ feat,
        const float* __restrict__ xs, const float* __restrict__ fs,
        const int* __restrict__ knn, int S, int N, int Cin,
        float* __restrict__ sx, float* __restrict__ sx2,
        float* __restrict__ sf, float* __restrict__ sf2) {
    int bs = blockIdx.x; int b = bs / S; int tid = threadIdx.x;
    __shared__ float lsx[32], lsx2[32], lsf[32], lsf2[32];
    __shared__ int nid[32];
    __shared__ float ctr[3];
    if (tid < 32) { lsx[tid] = 0; lsx2[tid] = 0; lsf[tid] = 0; lsf2[tid] = 0;
                    nid[tid] = knn[(size_t)bs * 32 + tid]; }
    if (tid < 3) ctr[tid] = xs[(size_t)bs * 3 + tid];
    __syncthreads();
    if (tid < 96) {
        int k = tid / 3, c = tid % 3;
        float d = x[((size_t)b * N + nid[k]) * 3 + c] - ctr[c];
        atomicAdd(&lsx[k], d); atomicAdd(&lsx2[k], d * d);
    }
    for (int i = tid; i < 32 * Cin; i += 128) {
        int k = i / Cin, c = i % Cin;
        float d = feat[((size_t)b * N + nid[k]) * Cin + c] - fs[(size_t)bs * Cin + c];
        atomicAdd(&lsf[k], d); atomicAdd(&lsf2[k], d * d);
    }
    __syncthreads();
    if (tid < 32) {
        atomicAdd(&sx[tid], lsx[tid]); atomicAdd(&sx2[tid], lsx2[tid]);
        atomicAdd(&sf[tid], lsf[tid]); atomicAdd(&sf2[tid], lsf2[tid]);
    }
}

__global__ void finalize_std_kernel(const float* __restrict__ sx, const float* __restrict__ sx2,
                                    const float* __restrict__ sf, const float* __restrict__ sf2,
                                    float nx, float nf,
                                    float* __restrict__ std_x, float* __restrict__ std_f) {
    int k = threadIdx.x;
    if (k < 32) {
        float vx = (sx2[k] - sx[k] * sx[k] / nx) / (nx - 1.f);
        std_x[k] = fmaxf(sqrtf(fmaxf(vx, 0.f)), 1e-5f);
        float vf = (sf2[k] - sf[k] * sf[k] / nf) / (nf - 1.f);
        std_f[k] = fmaxf(sqrtf(fmaxf(vf, 0.f)), 1e-5f);
    }
}

// -------------------------------- per-(b,c) stats of normalized xk (for stage embed)
__global__ void __launch_bounds__(96) gstd4_kernel(
        const float* __restrict__ x, const float* __restrict__ xs,
        const int* __restrict__ knn, const float* __restrict__ std_x,
        int S, int N, float* __restrict__ gb) {
    int bs = blockIdx.x; int b = bs / S; int tid = threadIdx.x;
    __shared__ int nid[32];
    __shared__ float ctr[3];
    __shared__ float acc[6];
    if (tid < 32) nid[tid] = knn[(size_t)bs * 32 + tid];
    if (tid < 3) ctr[tid] = xs[(size_t)bs * 3 + tid];
    if (tid < 6) acc[tid] = 0.f;
    __syncthreads();
    if (tid < 96) {
        int k = tid / 3, c = tid % 3;
        float v = (x[((size_t)b * N + nid[k]) * 3 + c] - ctr[c]) / std_x[k];
        atomicAdd(&acc[c * 2 + 0], v);
        atomicAdd(&acc[c * 2 + 1], v * v);
    }
    __syncthreads();
    if (tid < 6) {
        int c = tid >> 1, j = tid & 1;
        atomicAdd(&gb[((size_t)b * 3 + c) * 2 + j], acc[c * 2 + j]);
    }
}

// ------------------------ fused stage core: embed + (fk+pe)*pe + mean/max_K + GeLU
__global__ void stage_main_kernel(const float* __restrict__ x, const float* __restrict__ feat,
                                  const float* __restrict__ xs, const float* __restrict__ fs,
                                  const int* __restrict__ knn,
                                  const float* __restrict__ std_x, const float* __restrict__ std_f,
                                  const float* __restrict__ prm,
                                  int S, int N, int Cin, int fd,
                                  float* __restrict__ feat_out) {
    int bs = blockIdx.x; int b = bs / S; int j = threadIdx.x; int Cout = blockDim.x;
    __shared__ float xk[32][3];
    __shared__ int nid[32];
    __shared__ float sdf[32];
    if (j < 32) { nid[j] = knn[(size_t)bs * 32 + j]; sdf[j] = std_f[j]; }
    __syncthreads();
    if (j < 96) {
        int k = j / 3, c = j % 3;
        xk[k][c] = (x[((size_t)b * N + nid[k]) * 3 + c] - xs[(size_t)bs * 3 + c]) / std_x[k];
    }
    __syncthreads();
    float asig = prm[0], blend = prm[1];
    int f = (j * (3 * fd - 1)) / (Cout - 1);
    int c = f / fd, t = f % fd;
    float fv = -1.f + 2.f * (float)(t + 1) / (float)(fd + 1);
    float fsj = fs[(size_t)bs * Cin + (j < Cin ? j : j - Cin)];
    float accm = 0.f, accx = -FINF;
    for (int k = 0; k < 32; ++k) {
        float z = (xk[k][c] - fv) / (asig + 1e-6f);
        float pe = blend * expf(-0.5f * z * z) + (1.f - blend) * cosf(z);
        float fkv = (j < Cin) ? (feat[((size_t)b * N + nid[k]) * Cin + j] - fsj) / sdf[k]
                              : fsj;
        float fw = (fkv + pe) * pe;
        accm += fw; accx = fmaxf(accx, fw);
    }
    float g = accm * (1.f / 32.f) + accx;
    feat_out[(size_t)bs * Cout + j] = 0.5f * g * (1.f + erff(g * 0.70710678118654752f));
}

// ------------------------------------------- stage output: max_S and mean_S
__global__ void stage_out_kernel(const float* __restrict__ feat, int S, int Cout, int off,
                                 float* __restrict__ out, int outStride) {
    int bj = blockIdx.x; int b = bj / Cout, j = bj % Cout; int tid = threadIdx.x;
    __shared__ float rs[256], rm[256];
    float sm = 0.f, mx = -FINF;
    for (int s = tid; s < S; s += 256) {
        float v = feat[((size_t)b * S + s) * Cout + j];
        sm += v; mx = fmaxf(mx, v);
    }
    rs[tid] = sm; rm[tid] = mx; __syncthreads();
    for (int st = 128; st; st >>= 1) {
        if (tid < st) { rs[tid] += rs[tid + st]; rm[tid] = fmaxf(rm[tid], rm[tid + st]); }
        __syncthreads();
    }
    if (tid == 0) {
        out[(size_t)b * outStride + off + j] = rm[0];
        out[(size_t)b * outStride + off + Cout + j] = rs[0] / (float)S;
    }
}

// ================================================================= host driver
extern "C" void kernel_launch(void* const* d_in, const int* in_sizes, int n_in,
                              void* d_out, int out_size, void* d_ws, size_t ws_size,
                              hipStream_t stream) {
    (void)in_sizes; (void)n_in; (void)out_size; (void)ws_size;
    const float* xin = (const float*)d_in[0];   // [8, 4096, 3] f32
    float* out = (float*)d_out;                  // [8, 3840]  f32

    const int B = 8;
    static const int Ss[4]   = {2048, 1024, 512, 256};
    static const int Ns[4]   = {4096, 2048, 1024, 512};
    static const int Cins[4] = {64, 128, 256, 512};
    static const int fds[4]  = {43, 86, 171, 342};   // ceil(2*Cin/3)
    static const int offs[4] = {0, 256, 768, 1792};
    const int OUTSTRIDE = 3840;

    // workspace carve-out (aligned 256B)
    char* w = (char*)d_ws; size_t o = 0;
    auto take = [&](size_t nbytes) -> void* {
        o = (o + 255) & ~(size_t)255; void* p = w + o; o += nbytes; return p;
    };
    float* featA = (float*)take((size_t)B * 4096 * 64 * 4);  // all stage feats are same size
    float* featB = (float*)take((size_t)B * 4096 * 64 * 4);
    float* fsBuf = (float*)take((size_t)B * 2048 * 64 * 4);  // S*Cin constant across stages
    float* xsA   = (float*)take((size_t)B * 2048 * 3 * 4);
    float* xsB   = (float*)take((size_t)B * 2048 * 3 * 4);
    int*   fps   = (int*)  take((size_t)B * 2048 * 4);
    int*   knn   = (int*)  take((size_t)B * 2048 * 32 * 4);
    float* stats = (float*)take(1024 * 4);
    float *sx = stats, *sx2 = stats + 32, *sf = stats + 64, *sf2 = stats + 96,
          *gb = stats + 128, *stdx = stats + 176, *stdf = stats + 208,
          *prmI = stats + 240, *prmS = stats + 244;

    // ---- initial adaptive embed (dim 64) on raw xyz ----
    colstats_kernel<<<dim3(B * 3), dim3(256), 0, stream>>>(xin, 4096, gb);
    params_kernel<<<1, 1, 0, stream>>>(gb, 24, 4096.f, prmI);
    {
        size_t tot = (size_t)B * 4096 * 64;
        init_embed_kernel<<<(unsigned)((tot + 255) / 256), 256, 0, stream>>>(xin, prmI, featA, tot);
    }

    const float* x_cur = xin;
    float* feat_cur = featA;
    float* feat_nxt = featB;

    for (int st = 0; st < 4; ++st) {
        const int S = Ss[st], N = Ns[st], Cin = Cins[st], Cout = 2 * Cin, fd = fds[st];
        float* xs_buf = (st & 1) ? xsB : xsA;

        // (a) furthest point sampling (one block per batch, LDS-resident)
        fps_kernel<<<dim3(B), dim3(256), 0, stream>>>(x_cur, fps, N, S);

        // (b) gather sampled coords / feats
        {
            size_t tot = (size_t)B * S * 3;
            gather_kernel<<<(unsigned)((tot + 255) / 256), 256, 0, stream>>>(x_cur, fps, xs_buf, S, 3, N, tot);
            tot = (size_t)B * S * Cin;
            gather_kernel<<<(unsigned)((tot + 255) / 256), 256, 0, stream>>>(feat_cur, fps, fsBuf, S, Cin, N, tot);
        }

        // (c) KNN: WMMA f32 16x16x4 distances + streaming top-32
        knn_kernel<<<dim3(S / 16, B), dim3(512), 0, stream>>>(xs_buf, x_cur, knn, S, N);

        // (d) per-k std of coord/feature diffs (ddof=1, clipped at 1e-5)
        zero_kernel<<<1, 256, 0, stream>>>(stats, 176);  // sx..sf2 + gb
        diffstats_kernel<<<dim3(B * S), dim3(128), 0, stream>>>(x_cur, feat_cur, xs_buf, fsBuf,
                                                                knn, S, N, Cin, sx, sx2, sf, sf2);
        finalize_std_kernel<<<1, 32, 0, stream>>>(sx, sx2, sf, sf2,
                                                  (float)((size_t)B * S * 3),
                                                  (float)((size_t)B * S * Cin), stdx, stdf);

        // (e) embed params from normalized xk stats
        gstd4_kernel<<<dim3(B * S), dim3(96), 0, stream>>>(x_cur, xs_buf, knn, stdx, S, N, gb);
        params_kernel<<<1, 1, 0, stream>>>(gb, 24, (float)(S * 32), prmS);

        // (f) fused core: pe, (fk+pe)*pe, mean/max over K, GeLU
        stage_main_kernel<<<dim3(B * S), dim3(Cout), 0, stream>>>(x_cur, feat_cur, xs_buf, fsBuf,
                                                                  knn, stdx, stdf, prmS,
                                                                  S, N, Cin, fd, feat_nxt);

        // (g) stage output: [max_S | mean_S]
        stage_out_kernel<<<dim3(B * Cout), dim3(256), 0, stream>>>(feat_nxt, S, Cout, offs[st],
                                                                   out, OUTSTRIDE);

        // rotate buffers for next stage
        x_cur = xs_buf;
        float* tmp = feat_cur; feat_cur = feat_nxt;
        feat_nxt = (tmp == featA) ? featA : featB;  // reuse the other buffer
        feat_nxt = (feat_cur == featA) ? featB : featA;
    }
}